// qpNet_54219667144895
// MI455X (gfx1250) — compile-verified
//
#include <hip/hip_runtime.h>

typedef __attribute__((ext_vector_type(2))) float v2f;
typedef __attribute__((ext_vector_type(8))) float v8f;

#define NEG_INV_EPS (-1000.0f)   // -1/eps, eps = 1e-3

// One wave handles a tile of 16 rows of x per iteration.
// h = x @ W^T + b computed with two chained V_WMMA_F32_16X16X4_F32 (K = 4 + 1),
// z = max(-h, -1000) stored with NT hint. Data is loaded directly in the
// WMMA A-fragment layout (no LDS staging needed).
__global__ __launch_bounds__(256)
void qp_wmma_kernel(const float* __restrict__ x,
                    const float* __restrict__ W,     // [5,5] row-major: W[j*5+k]
                    const float* __restrict__ bfc,   // [5]
                    float* __restrict__ z,
                    int numTiles)
{
    const int lane  = threadIdx.x & 31;
    const int wpb   = blockDim.x >> 5;
    const int wave  = blockIdx.x * wpb + (threadIdx.x >> 5);
    const int nwave = gridDim.x * wpb;
    const int half  = lane >> 4;     // 0: K-block {0,1}, 1: K-block {2,3}
    const int nIdx  = lane & 15;     // output column N / A-matrix row-in-tile

    // ---- B fragments (4x16 f32 = 2 VGPRs): V0 holds K=(half?2:0), V1 holds K=(half?3:1),
    //      column N = lane%16. Columns N>=5 are zero (padding). Built once per wave.
    float b1v0 = 0.0f, b1v1 = 0.0f, b2v0 = 0.0f, cBias = 0.0f;
    if (nIdx < 5) {
        const float* wrow = W + nIdx * 5;          // W[nIdx][k]
        b1v0  = wrow[half ? 2 : 0];
        b1v1  = wrow[half ? 3 : 1];
        b2v0  = half ? 0.0f : wrow[4];             // second WMMA: only K=0 row live
        cBias = bfc[nIdx];
    }
    const v2f B1 = { b1v0, b1v1 };
    const v2f B2 = { b2v0, 0.0f };
    const v8f C  = { cBias, cBias, cBias, cBias, cBias, cBias, cBias, cBias };

    for (int t = wave; t < numTiles; t += nwave) {   // wave-uniform loop: EXEC all-1s at WMMA
        const int r   = t << 4;
        const int row = r + nIdx;
        const float* xr = x + (size_t)row * 5;

        // A fragment (16x4 f32): V0 = K(half?2:0), V1 = K(half?3:1) of row (r + lane%16).
        float a0 = __builtin_nontemporal_load(xr + (half ? 2 : 0));
        float a1 = __builtin_nontemporal_load(xr + (half ? 3 : 1));
        float a4 = __builtin_nontemporal_load(xr + 4);
        const v2f A1 = { a0, a1 };
        const v2f A2 = { half ? 0.0f : a4, 0.0f };   // only K=0 column populated

        v8f D = __builtin_amdgcn_wmma_f32_16x16x4_f32(
                    false, A1, false, B1, (short)0, C, false, false);
        D     = __builtin_amdgcn_wmma_f32_16x16x4_f32(
                    false, A2, false, B2, (short)0, D, false, false);

        // D layout: element (M = v + 8*half, N = lane%16) in VGPR v. Store N < 5.
        if (nIdx < 5) {
            #pragma unroll
            for (int v = 0; v < 8; ++v) {
                const int m = v + (half << 3);
                const float zz = fmaxf(-D[v], NEG_INV_EPS);   // v_max_num_f32 w/ neg modifier
                __builtin_nontemporal_store(zz, z + (size_t)(r + m) * 5 + nIdx);
            }
        }
    }
}

// Scalar tail for B % 16 != 0 (not hit for B = 4194304; launched only if needed).
__global__ __launch_bounds__(256)
void qp_tail_kernel(const float* __restrict__ x,
                    const float* __restrict__ W,
                    const float* __restrict__ bfc,
                    float* __restrict__ z,
                    int startRow, int numRows)
{
    int row = startRow + blockIdx.x * blockDim.x + threadIdx.x;
    if (row >= numRows) return;
    const float* xr = x + (size_t)row * 5;
    float xv[5];
    #pragma unroll
    for (int k = 0; k < 5; ++k) xv[k] = xr[k];
    #pragma unroll
    for (int j = 0; j < 5; ++j) {
        float h = bfc[j];
        #pragma unroll
        for (int k = 0; k < 5; ++k) h = fmaf(xv[k], W[j * 5 + k], h);
        z[(size_t)row * 5 + j] = fmaxf(-h, NEG_INV_EPS);
    }
}

extern "C" void kernel_launch(void* const* d_in, const int* in_sizes, int n_in,
                              void* d_out, int out_size, void* d_ws, size_t ws_size,
                              hipStream_t stream)
{
    const float* x   = (const float*)d_in[0];
    const float* W   = (const float*)d_in[1];
    const float* bfc = (const float*)d_in[2];
    float* z = (float*)d_out;

    const int B        = in_sizes[0] / 5;
    const int numTiles = B >> 4;           // 16 rows per WMMA tile

    if (numTiles > 0) {
        // 8 waves/block, target ~8 tiles per wave via grid-stride loop.
        int blocks = (numTiles + 63) / 64;
        if (blocks < 1) blocks = 1;
        qp_wmma_kernel<<<blocks, 256, 0, stream>>>(x, W, bfc, z, numTiles);
    }

    const int tailStart = numTiles << 4;
    const int tailRows  = B - tailStart;
    if (tailRows > 0) {
        int tb = (tailRows + 255) / 256;
        qp_tail_kernel<<<tb, 256, 0, stream>>>(x, W, bfc, z, tailStart, B);
    }
}